// Block_6579889898195
// MI455X (gfx1250) — compile-verified
//
#include <hip/hip_runtime.h>
#include <math.h>

typedef __attribute__((ext_vector_type(2))) float v2f;
typedef __attribute__((ext_vector_type(8))) float v8f;

// ---------------- constants ----------------
// B=64, T=4096, C=64
// level 0: L=2048, out_len=3069, pool_len=1535, nE=513, N=64*3069=196416, xn off=1 stride=2
// level 1: L=1024, out_len=1533, pool_len= 767, nE=257, N=64*1533= 98112, xn off=3 stride=4

// ---------------- global mean / std ----------------
__global__ void k_reduce_partial(const float* __restrict__ x, double* __restrict__ part) {
  __shared__ double s1[256], s2[256];
  const size_t n = 16777216ull; // 64*4096*64
  double a = 0.0, b = 0.0;
  for (size_t i = (size_t)blockIdx.x * blockDim.x + threadIdx.x; i < n;
       i += (size_t)gridDim.x * blockDim.x) {
    double t = (double)x[i];
    a += t; b += t * t;
  }
  s1[threadIdx.x] = a; s2[threadIdx.x] = b;
  __syncthreads();
  for (int off = 128; off > 0; off >>= 1) {
    if (threadIdx.x < (unsigned)off) {
      s1[threadIdx.x] += s1[threadIdx.x + off];
      s2[threadIdx.x] += s2[threadIdx.x + off];
    }
    __syncthreads();
  }
  if (threadIdx.x == 0) { part[blockIdx.x] = s1[0]; part[256 + blockIdx.x] = s2[0]; }
}

__global__ void k_finalize(const double* __restrict__ part, float* __restrict__ scal) {
  __shared__ double s1[256], s2[256];
  s1[threadIdx.x] = part[threadIdx.x];
  s2[threadIdx.x] = part[256 + threadIdx.x];
  __syncthreads();
  for (int off = 128; off > 0; off >>= 1) {
    if (threadIdx.x < (unsigned)off) {
      s1[threadIdx.x] += s1[threadIdx.x + off];
      s2[threadIdx.x] += s2[threadIdx.x + off];
    }
    __syncthreads();
  }
  if (threadIdx.x == 0) {
    double n = 16777216.0;
    double mean = s1[0] / n;
    double var = (s2[0] - s1[0] * s1[0] / n) / (n - 1.0); // ddof=1
    scal[0] = (float)(1.0 / mean);
    scal[1] = (float)(1.0 / sqrt(var));
  }
}

// ---------------- weight norm: W[co][ci*3+k] = g[co]/||v[co]|| * v ----------------
__global__ void k_weight_norm(const float* __restrict__ v, const float* __restrict__ g,
                              float* __restrict__ Wm) {
  int co = blockIdx.x;        // 64 blocks
  int tid = threadIdx.x;      // 64 threads
  __shared__ float red[64];
  __shared__ float scale;
  float a = 0.f;
  for (int i = tid; i < 192; i += 64) { float t = v[co * 192 + i]; a += t * t; }
  red[tid] = a;
  __syncthreads();
  for (int off = 32; off > 0; off >>= 1) {
    if (tid < off) red[tid] += red[tid + off];
    __syncthreads();
  }
  if (tid == 0) scale = g[co] / sqrtf(red[0]);
  __syncthreads();
  for (int i = tid; i < 192; i += 64) Wm[co * 192 + i] = scale * v[co * 192 + i];
}

// ---------------- sew-up indices (exact replica of _sew_up_indices state machine) ----------------
__global__ void k_sew(int* __restrict__ si) {
  if (blockIdx.x != 0 || threadIdx.x != 0) return;
  const int a_len = 2048, b_len = 4096;
  int skip_p = 1, skip_s = 1, skip_d = 2, skip_n = 3;
  bool skip_t = false;
  int cnt_a = 0, cnt_b = 0, w = 0;
  while (cnt_a < a_len) {
    if (cnt_a == 0) {
      int pv = cnt_b;
      for (int z = 0; z < skip_p; ++z) { si[w++] = a_len + pv; cnt_b++; }
      si[w++] = 0; cnt_a++;            // cur_layer==1 (odd)
      if (cnt_a == 0) cnt_a = 1;
      continue;
    }
    for (int z = 0; z < skip_n; ++z) {
      if (cnt_b >= b_len) break;
      si[w++] = a_len + cnt_b; cnt_b++;
    }
    if (!skip_t) {
      for (int z = 0; z < skip_s; ++z) {
        if (cnt_b >= b_len) break;
        si[w++] = a_len + cnt_b; cnt_b++;
        skip_t = true;
      }
    } else {
      for (int j = 0; j < skip_d; ++j) {
        for (int z = 0; z < skip_s; ++z) {
          if (cnt_b >= b_len) break;
          si[w++] = a_len + cnt_b; cnt_b++;
        }
        if (cnt_a >= a_len) break;
        si[w++] = cnt_a; cnt_a++;
        skip_t = false;
      }
      continue;
    }
    si[w++] = cnt_a; cnt_a++;
  }
  for (int j = cnt_b; j < b_len; ++j) si[w++] = a_len + j;
}

// ---------------- fused joined-gather, branchless (B operand of the conv GEMM) ------------
// joined[b][ci][p]: q=p/3, sel=p%3; prev=x[b, sS*q, ci]; xn=x[b, offX+sS*q, ci]
// sel0: 0.8*xn/mean + 0.2*prev ; sel1: prev ; sel2: 0.8*xn/std + 0.2*prev
__device__ __forceinline__ float bval(const float* __restrict__ x, int b_idx, int to, int kk,
                                      float im, float is, int sS, int offX) {
  unsigned ci = (unsigned)kk / 3u;
  unsigned k = (unsigned)kk - 3u * ci;
  unsigned p = 2u * (unsigned)to + 3u * k;   // stride 2, dilation 3
  unsigned q = p / 3u;
  unsigned sel = p - 3u * q;
  size_t base = ((size_t)b_idx * 4096 + (size_t)sS * q) * 64 + (size_t)ci;
  float pv = x[base];
  float xn = x[base + (size_t)offX * 64];
  float sc = (sel == 0u) ? im : is;
  float r = 0.8f * sc * xn + 0.2f * pv;
  return (sel == 1u) ? pv : r;
}

// ---------------- WMMA fp32 conv-as-GEMM ----------------
// Block = 128 threads (4 waves). Block owns one 16-wide N-tile; wave w owns M-tile w.
// The 192x16 B panel is staged once into LDS, pair-interleaved so each lane's per-step
// fragment {B[k0][n], B[k0+1][n]} is a single aligned ds_load_b64 (k0 always even).
__global__ void __launch_bounds__(128)
k_gemm_conv(const float* __restrict__ x, const float* __restrict__ Wm,
            const float* __restrict__ bias, const float* __restrict__ scal,
            float* __restrict__ O, int out_len, int N, int sS, int offX) {
  __shared__ float Bs[96 * 16 * 2];            // [pair=kk>>1][n][kk&1], 12 KB
  const int tid = threadIdx.x;
  const int ntile = blockIdx.x;

  // ---- stage B panel: 3072 elements, 24 per thread; nn fixed per thread ----
  {
    const int nn = tid & 15;
    const int n = ntile * 16 + nn;
    const int b_idx = n / out_len;
    const int to = n - b_idx * out_len;
    const float im = scal[0], is = scal[1];
    const int kk0 = tid >> 4;                  // 0..7
#pragma unroll
    for (int j = 0; j < 24; ++j) {
      int kk = kk0 + 8 * j;                    // covers 0..191
      float v = bval(x, b_idx, to, kk, im, is, sS, offX);
      Bs[((kk >> 1) * 16 + nn) * 2 + (kk & 1)] = v;
    }
  }
  __syncthreads();

  // ---- per-wave 16x16 WMMA tile ----
  const int lane = tid & 31;
  const int half = lane >> 4;
  const int l15 = lane & 15;
  const int mtile = tid >> 5;                  // wave id = M tile
  const int m = mtile * 16 + l15;              // A: lane-striped M
  const int n = ntile * 16 + l15;              // B/C/D: lane-striped N

  v8f acc = {0.f, 0.f, 0.f, 0.f, 0.f, 0.f, 0.f, 0.f};
  const float* wrow = Wm + (size_t)m * 192;    // row of normalized weights
  const float* brow = Bs + (size_t)l15 * 2;
#pragma unroll 4
  for (int kb = 0; kb < 48; ++kb) {            // K=192 in steps of 4
    int k0 = kb * 4 + 2 * half;                // VGPR0 = K0/K2 halves, VGPR1 = K1/K3
    v2f a = *(const v2f*)(wrow + k0);          // {W[m][k0], W[m][k0+1]} (8B aligned)
    v2f bv = *(const v2f*)(brow + (k0 >> 1) * 32); // {B[k0][n], B[k0+1][n]}
    acc = __builtin_amdgcn_wmma_f32_16x16x4_f32(false, a, false, bv, (short)0, acc,
                                                false, false);
  }
#pragma unroll
  for (int r = 0; r < 8; ++r) {                // C/D: VGPR r -> M = r + 8*half, N = l15
    int co = mtile * 16 + r + 8 * half;
    O[(size_t)co * N + n] = acc[r] + bias[co];
  }
}

// ---------------- maxpool(3,s2,pad1) -> elu -> exp(-e^2/2) ----------------
__global__ void k_pool_act(const float* __restrict__ O, float* __restrict__ Y,
                           int out_len, int plen, int N) {
  int idx = blockIdx.x * blockDim.x + threadIdx.x;
  int total = 64 * 64 * plen;
  if (idx >= total) return;
  int tp = idx % plen;
  int b = (idx / plen) & 63;
  int c = idx / (plen * 64);
  size_t base = (size_t)c * N + (size_t)b * out_len;
  float m = -INFINITY;
  for (int dt = -1; dt <= 1; ++dt) {
    int to = 2 * tp + dt;
    if (to >= 0 && to < out_len) m = fmaxf(m, O[base + to]);
  }
  float e = m > 0.f ? m : (expf(m) - 1.f);
  Y[idx] = expf(-0.5f * e * e);
}

// ---------------- batchnorm stats over dpad-stretched signal (weighted) ----------------
__global__ void k_bn_stats(const float* __restrict__ Y, float* __restrict__ stats,
                           int plen, int L, int nE) {
  int c = blockIdx.x;   // 64 blocks, 256 threads
  __shared__ double s1[256], s2[256];
  double a1 = 0.0, a2 = 0.0;
  int per = 64 * plen;
  for (int i = threadIdx.x; i < per; i += 256) {
    int tp = i % plen;
    double y = Y[(size_t)c * per + i];
    double w = (((tp & 1) == 0) && tp <= 2 * nE - 2) ? 2.0 : 1.0; // duplicated evens
    a1 += w * y; a2 += w * y * y;
  }
  s1[threadIdx.x] = a1; s2[threadIdx.x] = a2;
  __syncthreads();
  for (int off = 128; off > 0; off >>= 1) {
    if (threadIdx.x < (unsigned)off) {
      s1[threadIdx.x] += s1[threadIdx.x + off];
      s2[threadIdx.x] += s2[threadIdx.x + off];
    }
    __syncthreads();
  }
  if (threadIdx.x == 0) {
    double n = 64.0 * (double)L;
    double m = s1[0] / n;
    double var = s2[0] / n - m * m;
    stats[c] = (float)m;
    stats[64 + c] = (float)(1.0 / sqrt(var + 1e-5));
  }
}

// ---------------- final assembly: node-out + interleave + sew gathers ----------------
__device__ __forceinline__ float node_out(const float* __restrict__ x, const float* __restrict__ Y,
                                          float mu, float rv, float ga, float be,
                                          int b, int c, int t, int sS, int offX, int nE, int plen) {
  int thr = 3 * nE - 1;
  int tp = (t < thr) ? (2 * (t / 3) + ((t % 3) == 2 ? 1 : 0)) : (2 * nE - 1 + (t - thr));
  float P = Y[((size_t)c * 64 + b) * plen + tp];
  float bn = ga * (P - mu) * rv + be;
  float e = bn > 0.f ? bn : (expf(bn) - 1.f);
  float xn = x[((size_t)b * 4096 + offX + (size_t)sS * t) * 64 + c];
  return xn + e;
}

__global__ void k_final(const float* __restrict__ x, const float* __restrict__ Y0,
                        const float* __restrict__ Y1, const float* __restrict__ stats,
                        const float* __restrict__ gamma, const float* __restrict__ beta,
                        const int* __restrict__ si, float* __restrict__ out) {
  size_t idx = (size_t)blockIdx.x * blockDim.x + threadIdx.x;
  const size_t total = (size_t)64 * 64 * 6144;
  if (idx >= total) return;
  int v = (int)(idx % 6144);
  int c = (int)((idx / 6144) & 63);
  int b = (int)(idx / (6144 * 64));
  int s = si[v];
  float val;
  if (s < 2048) {                       // layer1 (interleave of nodes[2] and out1)
    int u = s;
    if ((u & 1) == 0) val = x[((size_t)b * 4096 + 2 * (size_t)u) * 64 + c];
    else val = node_out(x, Y1, stats[128 + c], stats[192 + c], gamma[c], beta[c],
                        b, c, (u - 1) >> 1, 4, 3, 257, 767);
  } else {                              // layer0 (interleave of nodes[0] and out0)
    int u = s - 2048;
    if ((u & 1) == 0) val = x[((size_t)b * 4096 + (size_t)u) * 64 + c];
    else val = node_out(x, Y0, stats[c], stats[64 + c], gamma[c], beta[c],
                        b, c, (u - 1) >> 1, 2, 1, 513, 1535);
  }
  out[idx] = val;
}

// ---------------- launch ----------------
extern "C" void kernel_launch(void* const* d_in, const int* in_sizes, int n_in,
                              void* d_out, int out_size, void* d_ws, size_t ws_size,
                              hipStream_t stream) {
  (void)in_sizes; (void)n_in; (void)out_size; (void)ws_size;
  const float* x      = (const float*)d_in[0];
  const float* conv_v = (const float*)d_in[1];
  const float* conv_g = (const float*)d_in[2];
  const float* conv_b = (const float*)d_in[3];
  const float* gamma  = (const float*)d_in[4];
  const float* beta   = (const float*)d_in[5];
  float* out = (float*)d_out;

  char* ws = (char*)d_ws;                           // ~88.1 MB used
  float*  scal  = (float*)(ws + 0);                 // inv_mean, inv_std
  double* part  = (double*)(ws + 256);              // 512 doubles
  float*  Wm    = (float*)(ws + 4608);              // 64*192
  float*  stats = (float*)(ws + 53760);             // mu0,rinv0,mu1,rinv1 (4*64)
  int*    si    = (int*)(ws + 54784);               // 6144 sew indices
  float*  Y0    = (float*)(ws + 79360);             // 64*64*1535
  float*  Y1    = (float*)(ws + 25228800);          // 64*64*767
  float*  O     = (float*)(ws + 37795328);          // 64*196416 (reused both levels)

  k_reduce_partial<<<256, 256, 0, stream>>>(x, part);
  k_finalize<<<1, 256, 0, stream>>>(part, scal);
  k_weight_norm<<<64, 64, 0, stream>>>(conv_v, conv_g, Wm);
  k_sew<<<1, 1, 0, stream>>>(si);

  // level 0: L=2048, out_len=3069, N=196416 (12276 N-tiles), pool_len=1535, nE=513
  k_gemm_conv<<<12276, 128, 0, stream>>>(x, Wm, conv_b, scal, O, 3069, 196416, 2, 1);
  k_pool_act<<<(6287360 + 255) / 256, 256, 0, stream>>>(O, Y0, 3069, 1535, 196416);
  k_bn_stats<<<64, 256, 0, stream>>>(Y0, stats, 1535, 2048, 513);

  // level 1: L=1024, out_len=1533, N=98112 (6132 N-tiles), pool_len=767, nE=257
  k_gemm_conv<<<6132, 128, 0, stream>>>(x, Wm, conv_b, scal, O, 1533, 98112, 4, 3);
  k_pool_act<<<(3141632 + 255) / 256, 256, 0, stream>>>(O, Y1, 1533, 767, 98112);
  k_bn_stats<<<64, 256, 0, stream>>>(Y1, stats + 128, 767, 1024, 257);

  k_final<<<(25165824 + 255) / 256, 256, 0, stream>>>(x, Y0, Y1, stats, gamma, beta, si, out);
}